// TopologyGNN_70995809402913
// MI455X (gfx1250) — compile-verified
//
#include <hip/hip_runtime.h>

typedef float v2f __attribute__((ext_vector_type(2)));
typedef float v8f __attribute__((ext_vector_type(8)));

#define F1 128
#define F2 16
#define F3 8

__device__ __forceinline__ void atomAddF(float* p, float v) {
    unsafeAtomicAdd(p, v);   // lowers to global_atomic_add_f32 (non-returning)
}

// ---------------- init: deg=1.0 (self loop), agg1=0, out=0 ----------------
__global__ void k_init(float* __restrict__ deg, float* __restrict__ agg1,
                       float* __restrict__ out, int N) {
    int i = blockIdx.x * blockDim.x + threadIdx.x;
    if (i < N) deg[i] = 1.0f;
    if (i < N * F2) agg1[i] = 0.0f;
    if (i < N * F3) out[i] = 0.0f;
}

// ---------------- degree accumulation ----------------
__global__ void k_deg(const int* __restrict__ dst, float* __restrict__ deg, int E) {
    int e = blockIdx.x * blockDim.x + threadIdx.x;
    if (e >= E) return;
    atomAddF(&deg[dst[e]], 1.0f);
}

// ---------------- dinv = rsqrt(deg), in place ----------------
__global__ void k_dinv(float* __restrict__ deg, int N) {
    int i = blockIdx.x * blockDim.x + threadIdx.x;
    if (i < N) deg[i] = rsqrtf(deg[i]);
}

// ---------------- GEMM1: H1[N,16] = X[N,128] @ W1[128,16] via V_WMMA_F32_16X16X4_F32 ----------------
// One wave32 computes a 16x16 tile (16 node rows x 16 out features), K=128 in 32 steps.
__global__ void k_gemm1(const float* __restrict__ X, const float* __restrict__ W,
                        float* __restrict__ H, int N) {
    int wave = (int)((blockIdx.x * blockDim.x + threadIdx.x) >> 5);
    int lane = threadIdx.x & 31;
    int row0 = wave * 16;
    if (row0 >= N) return;                 // wave-uniform exit

    int m     = lane & 15;                 // A: M index / B: N index
    int khalf = (lane >> 4) << 1;          // 0 for lanes 0-15, 2 for lanes 16-31

    const float* xrow = X + (size_t)(row0 + m) * F1;
    v8f c = {};
#pragma unroll 4
    for (int k = 0; k < F1; k += 4) {
        v2f a, b;
        a.x = xrow[k + khalf];             // A[M=m, K=k+khalf]
        a.y = xrow[k + khalf + 1];         // A[M=m, K=k+khalf+1]
        b.x = W[(k + khalf) * F2 + m];     // B[K, N=m]
        b.y = W[(k + khalf + 1) * F2 + m];
        c = __builtin_amdgcn_wmma_f32_16x16x4_f32(false, a, false, b,
                                                  (short)0, c, false, false);
    }
    // D layout: lanes 0-15 -> M=0..7 (vgpr 0..7), N=lane; lanes 16-31 -> M=8..15, N=lane-16
    int col   = lane & 15;
    int rbase = row0 + ((lane >> 4) << 3);
#pragma unroll
    for (int v = 0; v < 8; v++)
        H[(size_t)(rbase + v) * F2 + col] = c[v];
}

// ---------------- edge scatter layer1: agg1[dst] += h1[src] * dinv[src]*dinv[dst] ----------------
// 4 threads per edge, float4 per thread (16 features)
__global__ void k_scatter1(const int* __restrict__ src, const int* __restrict__ dst,
                           const float* __restrict__ H, const float* __restrict__ dinv,
                           float* __restrict__ agg, int E) {
    long long gid = (long long)blockIdx.x * blockDim.x + threadIdx.x;
    int e = (int)(gid >> 2), q = (int)(gid & 3);
    if (e >= E) return;
    int s = src[e], d = dst[e];
    float norm = dinv[s] * dinv[d];
    float4 h = ((const float4*)(H + (size_t)s * F2))[q];
    float* a = agg + (size_t)d * F2 + q * 4;
    atomAddF(a + 0, h.x * norm);
    atomAddF(a + 1, h.y * norm);
    atomAddF(a + 2, h.z * norm);
    atomAddF(a + 3, h.w * norm);
}

// ---------------- finalize1: z1 = relu(agg1 + h1*dinv^2 + b1), in place over agg1 ----------------
__global__ void k_fin1(float* __restrict__ agg, const float* __restrict__ H,
                       const float* __restrict__ dinv, const float* __restrict__ b,
                       int N) {
    int idx = blockIdx.x * blockDim.x + threadIdx.x;
    if (idx >= N * F2) return;
    int i = idx >> 4, f = idx & 15;
    float di = dinv[i];
    float v = agg[idx] + H[idx] * di * di + b[f];
    agg[idx] = v > 0.0f ? v : 0.0f;
}

// ---------------- GEMM2: H2[N,8] = Z1[N,16] @ W2[16,8] via WMMA (N padded to 16) ----------------
__global__ void k_gemm2(const float* __restrict__ Z, const float* __restrict__ W,
                        float* __restrict__ H, int N) {
    int wave = (int)((blockIdx.x * blockDim.x + threadIdx.x) >> 5);
    int lane = threadIdx.x & 31;
    int row0 = wave * 16;
    if (row0 >= N) return;

    int m     = lane & 15;
    int khalf = (lane >> 4) << 1;
    const float* zrow = Z + (size_t)(row0 + m) * F2;
    v8f c = {};
#pragma unroll
    for (int k = 0; k < F2; k += 4) {
        v2f a, b;
        a.x = zrow[k + khalf];
        a.y = zrow[k + khalf + 1];
        b.x = (m < F3) ? W[(k + khalf) * F3 + m] : 0.0f;
        b.y = (m < F3) ? W[(k + khalf + 1) * F3 + m] : 0.0f;
        c = __builtin_amdgcn_wmma_f32_16x16x4_f32(false, a, false, b,
                                                  (short)0, c, false, false);
    }
    int col   = lane & 15;
    int rbase = row0 + ((lane >> 4) << 3);
    if (col < F3) {
#pragma unroll
        for (int v = 0; v < 8; v++)
            H[(size_t)(rbase + v) * F3 + col] = c[v];
    }
}

// ---------------- edge scatter layer2: out[dst] += h2[src]*norm ----------------
// 2 threads per edge, float4 per thread (8 features)
__global__ void k_scatter2(const int* __restrict__ src, const int* __restrict__ dst,
                           const float* __restrict__ H, const float* __restrict__ dinv,
                           float* __restrict__ out, int E) {
    long long gid = (long long)blockIdx.x * blockDim.x + threadIdx.x;
    int e = (int)(gid >> 1), q = (int)(gid & 1);
    if (e >= E) return;
    int s = src[e], d = dst[e];
    float norm = dinv[s] * dinv[d];
    float4 h = ((const float4*)(H + (size_t)s * F3))[q];
    float* a = out + (size_t)d * F3 + q * 4;
    atomAddF(a + 0, h.x * norm);
    atomAddF(a + 1, h.y * norm);
    atomAddF(a + 2, h.z * norm);
    atomAddF(a + 3, h.w * norm);
}

// ---------------- finalize2: out += h2*dinv^2 + b2 ----------------
__global__ void k_fin2(float* __restrict__ out, const float* __restrict__ H,
                       const float* __restrict__ dinv, const float* __restrict__ b,
                       int N) {
    int idx = blockIdx.x * blockDim.x + threadIdx.x;
    if (idx >= N * F3) return;
    int i = idx >> 3, f = idx & 7;
    float di = dinv[i];
    out[idx] += H[idx] * di * di + b[f];
}

extern "C" void kernel_launch(void* const* d_in, const int* in_sizes, int n_in,
                              void* d_out, int out_size, void* d_ws, size_t ws_size,
                              hipStream_t stream) {
    const float* X  = (const float*)d_in[0];
    const int*   EI = (const int*)d_in[1];
    const float* W1 = (const float*)d_in[2];
    const float* b1 = (const float*)d_in[3];
    const float* W2 = (const float*)d_in[4];
    const float* b2 = (const float*)d_in[5];

    const int N = in_sizes[0] / F1;     // 100000
    const int E = in_sizes[1] / 2;      // 3200000
    const int* src = EI;
    const int* dst = EI + E;

    // workspace carve-out (256B aligned): dinv | h1 | agg1 | h2
    char* ws = (char*)d_ws;
    auto carve = [&](size_t bytes) {
        char* p = ws;
        ws += (bytes + 255) & ~(size_t)255;
        return p;
    };
    float* dinv = (float*)carve((size_t)N * 4);
    float* h1   = (float*)carve((size_t)N * F2 * 4);
    float* agg1 = (float*)carve((size_t)N * F2 * 4);
    float* h2   = (float*)carve((size_t)N * F3 * 4);
    float* out  = (float*)d_out;

    const int T = 256;
    int tiles  = N / 16;                       // N divisible by 16
    int gblk   = (tiles + 7) / 8;              // 8 waves (tiles) per 256-thread block

    k_init<<<(N * F2 + T - 1) / T, T, 0, stream>>>(dinv, agg1, out, N);
    k_deg<<<(E + T - 1) / T, T, 0, stream>>>(dst, dinv, E);
    k_dinv<<<(N + T - 1) / T, T, 0, stream>>>(dinv, N);

    k_gemm1<<<gblk, T, 0, stream>>>(X, W1, h1, N);

    long long n1 = (long long)E * 4;
    k_scatter1<<<(int)((n1 + T - 1) / T), T, 0, stream>>>(src, dst, h1, dinv, agg1, E);
    k_fin1<<<(N * F2 + T - 1) / T, T, 0, stream>>>(agg1, h1, dinv, b1, N);

    k_gemm2<<<gblk, T, 0, stream>>>(agg1, W2, h2, N);

    long long n2 = (long long)E * 2;
    k_scatter2<<<(int)((n2 + T - 1) / T), T, 0, stream>>>(src, dst, h2, dinv, out, E);
    k_fin2<<<(N * F3 + T - 1) / T, T, 0, stream>>>(out, h2, dinv, b2, N);
}